// YOLOLoss_75110388072502
// MI455X (gfx1250) — compile-verified
//
#include <hip/hip_runtime.h>
#include <math.h>

// ---------------------------------------------------------------------------
// YOLOv5-style loss for MI455X (gfx1250, wave32).
//
// Dense part: sum of softplus(objectness) over all B*A*H*W cells per scale.
//   -> only the objectness channel planes are read (25.8 MB total vs 155 MB
//      naive); bandwidth-bound floor = 25.8MB / 23.3 TB/s ~ 1.1us.
//   -> softplus computed in base-2 with raw v_exp_f32/v_log_f32 builtins
//      (2 TRANS + ~5 VALU ops/element, no OCML fixup code), keeping ALU
//      work under the memory roofline.
//   -> accumulation runs on the matrix pipe: V_WMMA_F32_16X16X4_F32 with
//      B = ones; each b128 load feeds two WMMAs; 4 independent f32
//      accumulators; every value lands in 16 columns -> divide by 16.
// Sparse part: 2048 boxes/scale -> anchor match + bitmask dedup + 5-float
//   gather; negatives = (all-cells sum) - (positive-cell softplus).
// ---------------------------------------------------------------------------

typedef float v2f __attribute__((ext_vector_type(2)));
typedef float v4f __attribute__((ext_vector_type(4)));
typedef float v8f __attribute__((ext_vector_type(8)));

#define BATCH 64
#define NBOX  32
#define NA    3
#define NCH   6   // nc + 5, nc = 1
#define LBOX  0.05f
#define LOBJ  1.0f

#define LOG2E 1.44269504088896340736f
#define LN2   0.69314718055994530942f

// workspace layout (32-bit words)
#define ACC_WORDS 32                      // 3 scales x 8 floats (padded)
#define BM0_WORDS (BATCH * NA * 160 * 160 / 32)   // 153600
#define BM1_WORDS (BATCH * NA *  80 *  80 / 32)   //  38400
#define BM2_WORDS (BATCH * NA *  40 *  40 / 32)   //   9600
#define BM0_OFF   (ACC_WORDS)
#define BM1_OFF   (BM0_OFF + BM0_WORDS)
#define BM2_OFF   (BM1_OFF + BM1_WORDS)
#define WS_WORDS  (BM2_OFF + BM2_WORDS)

// Precise softplus (sparse path only): logaddexp(x,0)
__device__ __forceinline__ float softplus_precise(float x) {
  return fmaxf(x, 0.0f) + log1pf(expf(-fabsf(x)));
}

// Fast softplus (dense path), base-2 hardware transcendentals:
//   e = exp2(-|x|*log2e) in (0,1]  -> 1+e in [1,2], log well-conditioned.
//   softplus(x) = max(x,0) + ln2 * log2(1+e).
// Inputs finite by construction: no special-case fixups needed.
__device__ __forceinline__ float softplus_fast(float x) {
  const float e = __builtin_amdgcn_exp2f(-fabsf(x) * LOG2E);  // v_exp_f32
  const float l = __builtin_amdgcn_logf(1.0f + e);            // v_log_f32
  return fmaxf(x, 0.0f) + LN2 * l;
}

__device__ __forceinline__ v8f wmma_acc(v2f a, v2f ones, v8f c) {
  return __builtin_amdgcn_wmma_f32_16x16x4_f32(false, a, false, ones,
                                               (short)0, c, false, false);
}

__global__ void zero_ws_kernel(unsigned* __restrict__ ws, int nWords) {
  for (int i = blockIdx.x * blockDim.x + threadIdx.x; i < nWords;
       i += gridDim.x * blockDim.x)
    ws[i] = 0u;
}

// ---- dense reduction, b128 path (HW % 128 == 0): 128 elements per chunk ----
template <int G>
__global__ void obj_neg_sum_b128(const float* __restrict__ pred,
                                 float* __restrict__ accum,  // &acc[s*8+0]
                                 int nChunks) {
  constexpr unsigned HW  = (unsigned)G * (unsigned)G;
  constexpr unsigned CPP = HW / 128u;                // chunks per plane
  const int lane   = threadIdx.x & 31;
  const int wave   = blockIdx.x * (blockDim.x >> 5) + (threadIdx.x >> 5);
  const int nWaves = gridDim.x * (blockDim.x >> 5);
  const unsigned laneOff = (unsigned)lane * 4u;      // float4 per lane
  const v2f ones = {1.0f, 1.0f};
  v8f c0 = {}, c1 = {}, c2 = {}, c3 = {};

  auto base_of = [&](unsigned chunk) -> unsigned {
    const unsigned plane  = chunk / CPP;             // const divisor: magic mul
    const unsigned within = (chunk - plane * CPP) * 128u;
    const unsigned b = plane / NA;
    const unsigned a = plane - b * NA;
    return (b * (NA * NCH) + a * NCH + 4u) * HW + within + laneOff;
  };

  int chunk = wave;
  for (; chunk + nWaves < nChunks; chunk += 2 * nWaves) {
    const unsigned b0 = base_of((unsigned)chunk);
    const unsigned b1 = base_of((unsigned)(chunk + nWaves));
    const v4f x0 = *(const v4f*)(pred + b0);         // global_load_b128
    const v4f x1 = *(const v4f*)(pred + b1);
    __builtin_prefetch(pred + b1 + (size_t)nWaves * 128, 0, 3);
    v2f a0 = {softplus_fast(x0.x), softplus_fast(x0.y)};
    v2f a1 = {softplus_fast(x0.z), softplus_fast(x0.w)};
    v2f a2 = {softplus_fast(x1.x), softplus_fast(x1.y)};
    v2f a3 = {softplus_fast(x1.z), softplus_fast(x1.w)};
    c0 = wmma_acc(a0, ones, c0);
    c1 = wmma_acc(a1, ones, c1);
    c2 = wmma_acc(a2, ones, c2);
    c3 = wmma_acc(a3, ones, c3);
  }
  if (chunk < nChunks) {                             // tail (uniform per wave)
    const unsigned b0 = base_of((unsigned)chunk);
    const v4f x0 = *(const v4f*)(pred + b0);
    v2f a0 = {softplus_fast(x0.x), softplus_fast(x0.y)};
    v2f a1 = {softplus_fast(x0.z), softplus_fast(x0.w)};
    c0 = wmma_acc(a0, ones, c0);
    c1 = wmma_acc(a1, ones, c1);
  }

  float s = 0.0f;
#pragma unroll
  for (int i = 0; i < 8; ++i) s += (c0[i] + c1[i]) + (c2[i] + c3[i]);
#pragma unroll
  for (int m = 16; m >= 1; m >>= 1) s += __shfl_xor(s, m, 32);
  if (lane == 0) atomicAdd(accum, s * (1.0f / 16.0f));
}

// ---- dense reduction, b64 path (G = 40): 64 elements per chunk -------------
template <int G>
__global__ void obj_neg_sum_b64(const float* __restrict__ pred,
                                float* __restrict__ accum,
                                int nChunks) {
  constexpr unsigned HW  = (unsigned)G * (unsigned)G;
  constexpr unsigned CPP = HW / 64u;
  const int lane   = threadIdx.x & 31;
  const int wave   = blockIdx.x * (blockDim.x >> 5) + (threadIdx.x >> 5);
  const int nWaves = gridDim.x * (blockDim.x >> 5);
  const unsigned laneOff = (unsigned)lane * 2u;
  const v2f ones = {1.0f, 1.0f};
  v8f c0 = {}, c1 = {};

  auto base_of = [&](unsigned chunk) -> unsigned {
    const unsigned plane  = chunk / CPP;
    const unsigned within = (chunk - plane * CPP) * 64u;
    const unsigned b = plane / NA;
    const unsigned a = plane - b * NA;
    return (b * (NA * NCH) + a * NCH + 4u) * HW + within + laneOff;
  };

  int chunk = wave;
  for (; chunk + nWaves < nChunks; chunk += 2 * nWaves) {
    const unsigned b0 = base_of((unsigned)chunk);
    const unsigned b1 = base_of((unsigned)(chunk + nWaves));
    const v2f x0 = *(const v2f*)(pred + b0);
    const v2f x1 = *(const v2f*)(pred + b1);
    v2f a0 = {softplus_fast(x0.x), softplus_fast(x0.y)};
    v2f a1 = {softplus_fast(x1.x), softplus_fast(x1.y)};
    c0 = wmma_acc(a0, ones, c0);
    c1 = wmma_acc(a1, ones, c1);
  }
  if (chunk < nChunks) {
    const unsigned b0 = base_of((unsigned)chunk);
    const v2f x0 = *(const v2f*)(pred + b0);
    v2f a0 = {softplus_fast(x0.x), softplus_fast(x0.y)};
    c0 = wmma_acc(a0, ones, c0);
  }

  float s = 0.0f;
#pragma unroll
  for (int i = 0; i < 8; ++i) s += c0[i] + c1[i];
#pragma unroll
  for (int m = 16; m >= 1; m >>= 1) s += __shfl_xor(s, m, 32);
  if (lane == 0) atomicAdd(accum, s * (1.0f / 16.0f));
}

// ---- sparse positives: one thread per (b, n) box ---------------------------
__global__ void pos_kernel(const float* __restrict__ pred,
                           const float* __restrict__ boxes,
                           const float* __restrict__ anchors,   // 3x2 for scale
                           unsigned* __restrict__ bitmask,
                           float* __restrict__ acc,             // &acc[s*8]
                           int G) {
  const int t = blockIdx.x * blockDim.x + threadIdx.x;
  if (t >= BATCH * NBOX) return;
  const int b = t / NBOX;
  const int HW = G * G;

  const float* bx = boxes + (long)t * 4;
  const float gf = (float)G;
  const float gx = bx[0] * gf, gy = bx[1] * gf;
  const float gw = bx[2] * gf, gh = bx[3] * gf;
  const int gi = (int)gx, gj = (int)gy;   // truncation; boxes strictly inside

  // anchor match: argmax_a prod(min(r, 1/r)); first max wins (JAX argmax)
  int best = 0;
  float bestIou = -1.0f;
#pragma unroll
  for (int a = 0; a < NA; ++a) {
    const float rw = anchors[a * 2 + 0] / gw;
    const float rh = anchors[a * 2 + 1] / gh;
    const float iou = fminf(rw, 1.0f / rw) * fminf(rh, 1.0f / rh);
    if (iou > bestIou) { bestIou = iou; best = a; }
  }

  const int cell = ((b * NA + best) * G + gj) * G + gi;
  const unsigned bit = 1u << (cell & 31);
  const unsigned old = atomicOr(&bitmask[cell >> 5], bit);
  if (old & bit) return;   // duplicate (b, a, gj, gi): one writer wins

  const float tx = gx - (float)gi;
  const float ty = gy - (float)gj;
  const float tw = logf(gw / anchors[best * 2 + 0] + 1e-16f);
  const float th = logf(gh / anchors[best * 2 + 1] + 1e-16f);

  const long cbase = (long)(b * (NA * NCH) + best * NCH) * HW + (long)gj * G + gi;
  const float p0 = pred[cbase + 0L * HW];
  const float p1 = pred[cbase + 1L * HW];
  const float p2 = pred[cbase + 2L * HW];
  const float p3 = pred[cbase + 3L * HW];
  const float p4 = pred[cbase + 4L * HW];

  const float sq = (p0 - tx) * (p0 - tx) + (p1 - ty) * (p1 - ty) +
                   (p2 - tw) * (p2 - tw) + (p3 - th) * (p3 - th);

  atomicAdd(&acc[1], softplus_precise(-p4));  // positive objectness term
  atomicAdd(&acc[2], softplus_fast(p4));      // matches dense-sum formulation
  atomicAdd(&acc[3], sq);                     // box squared error
  atomicAdd(&acc[4], 1.0f);                   // distinct positive cell count
}

__global__ void finalize_kernel(const float* __restrict__ acc,
                                float* __restrict__ out) {
  const float totals[3] = {64.0f * 3 * 160 * 160,
                           64.0f * 3 * 80 * 80,
                           64.0f * 3 * 40 * 40};
  float loss = 0.0f;
#pragma unroll
  for (int s = 0; s < 3; ++s) {
    const float negAll     = acc[s * 8 + 0];
    const float posObj     = acc[s * 8 + 1];
    const float posCellNeg = acc[s * 8 + 2];
    const float sq         = acc[s * 8 + 3];
    const float npos       = acc[s * 8 + 4];
    const float nneg       = totals[s] - npos;
    loss += LBOX * (sq / npos) +
            LOBJ * (posObj / npos + (negAll - posCellNeg) / nneg);
  }
  out[0] = loss;
}

extern "C" void kernel_launch(void* const* d_in, const int* in_sizes, int n_in,
                              void* d_out, int out_size, void* d_ws, size_t ws_size,
                              hipStream_t stream) {
  (void)in_sizes; (void)n_in; (void)out_size; (void)ws_size;
  const float* pred0 = (const float*)d_in[0];
  const float* pred1 = (const float*)d_in[1];
  const float* pred2 = (const float*)d_in[2];
  const float* boxes   = (const float*)d_in[3];
  // d_in[4] = labels (unused, nc == 1)
  const float* anchors = (const float*)d_in[5];   // [3, 3, 2]

  unsigned* ws  = (unsigned*)d_ws;
  float*    acc = (float*)d_ws;                   // first ACC_WORDS words
  float*    out = (float*)d_out;

  // 1) zero accumulators + dedup bitmasks (fresh state every call)
  zero_ws_kernel<<<(WS_WORDS + 255) / 256, 256, 0, stream>>>(ws, WS_WORDS);

  // 2) dense objectness reduction per scale (WMMA f32 accumulate)
  obj_neg_sum_b128<160><<<960, 256, 0, stream>>>(
      pred0, acc + 0 * 8, BATCH * NA * 160 * 160 / 128);
  obj_neg_sum_b128<80><<<480, 256, 0, stream>>>(
      pred1, acc + 1 * 8, BATCH * NA * 80 * 80 / 128);
  obj_neg_sum_b64<40><<<120, 256, 0, stream>>>(
      pred2, acc + 2 * 8, BATCH * NA * 40 * 40 / 64);

  // 3) sparse positives per scale
  const float* preds[3] = {pred0, pred1, pred2};
  const int grids[3] = {160, 80, 40};
  const int bmOff[3] = {BM0_OFF, BM1_OFF, BM2_OFF};
  for (int s = 0; s < 3; ++s) {
    pos_kernel<<<(BATCH * NBOX + 255) / 256, 256, 0, stream>>>(
        preds[s], boxes, anchors + s * 6, ws + bmOff[s], acc + s * 8, grids[s]);
  }

  // 4) combine
  finalize_kernel<<<1, 1, 0, stream>>>(acc, out);
}